// Codebook_74818330296514
// MI455X (gfx1250) — compile-verified
//
#include <hip/hip_runtime.h>

typedef __attribute__((ext_vector_type(2))) float v2f;
typedef __attribute__((ext_vector_type(8))) float v8f;

#define N_ROWS 16384   // 8*2048 tokens
#define K_CODES 8192
#define DIM 32

// ---------------------------------------------------------------------------
// Kernel 1: L2-normalize z rows and codebook rows; also emit esq[k] = sum(en^2)
// One wave32 per row (DIM == 32 -> one element per lane).
// ---------------------------------------------------------------------------
__global__ __launch_bounds__(256) void k_normalize(const float* __restrict__ z,
                                                   const float* __restrict__ cb,
                                                   float* __restrict__ zn,
                                                   float* __restrict__ en,
                                                   float* __restrict__ esq) {
    int warp = (int)((blockIdx.x * blockDim.x + threadIdx.x) >> 5);
    int lane = (int)(threadIdx.x & 31);
    if (warp >= N_ROWS + K_CODES) return;
    bool isZ = warp < N_ROWS;
    const float* src = isZ ? (z + (size_t)warp * DIM)
                           : (cb + (size_t)(warp - N_ROWS) * DIM);
    float x = src[lane];
    float ss = x * x;
    #pragma unroll
    for (int m = 16; m >= 1; m >>= 1) ss += __shfl_xor(ss, m, 32);
    float nrm = sqrtf(ss);
    float inv = 1.0f / fmaxf(nrm, 1e-12f);
    float r = x * inv;
    if (isZ) {
        zn[(size_t)warp * DIM + lane] = r;
    } else {
        int row = warp - N_ROWS;
        en[(size_t)row * DIM + lane] = r;
        float rs = r * r;                 // recompute sum(en^2) like the reference
        #pragma unroll
        for (int m = 16; m >= 1; m >>= 1) rs += __shfl_xor(rs, m, 32);
        if (lane == 0) esq[row] = rs;
    }
}

// ---------------------------------------------------------------------------
// Kernel 2: fused fp32 WMMA GEMM + row-wise argmax of (dot - esq/2).
// Block = 256 threads (8 waves). Block owns one 16-row tile of zn;
// waves stride over the 512 codebook tiles of 16.
//
// The bias -0.5*esq[col] is folded INTO the GEMM as a 9th K-chunk:
//   A chunk 8 = constant column 1.0 (lanes 0-15 hold K={32,33} = {1,0};
//               lanes 16-31 hold K={34,35} = {0,0})
//   B chunk 8 = row of -0.5*esq[col]   (lanes 0-15: {bias, 0}; lanes 16-31: 0)
// so one single-accumulator chain of 9 v_wmma_f32_16x16x4_f32 starting from
// inline C=0 yields score = dot - esq/2 directly (no splat movs, no join adds).
// The lane-half predicate is applied via a hoisted multiplier (hscale) so the
// inner loop stays branch-free (no EXEC save/restore per tile).
//
// A (16x4 f32) layout: lanes 0-15 hold K={0,1}, lanes 16-31 hold K={2,3},
// row = lane%16. B (4x16 f32): same K split across lane halves, col = lane%16.
// C/D (16x16 f32): slot v, lanes 0-15 -> row v, lanes 16-31 -> row v+8,
// col = lane%16.
// ---------------------------------------------------------------------------
__global__ __launch_bounds__(256) void k_argmin(const float* __restrict__ zn,
                                                const float* __restrict__ en,
                                                const float* __restrict__ esq,
                                                int* __restrict__ bestIdx) {
    __shared__ float sS[8][16];
    __shared__ int   sI[8][16];

    const int tile = (int)blockIdx.x;          // 0..1023 row tiles
    // wave id is uniform within a wave: force it scalar so the k-loop is SALU
    const int wave = __builtin_amdgcn_readfirstlane((int)(threadIdx.x >> 5));
    const int lane = (int)(threadIdx.x & 31);
    const int l15  = lane & 15;
    const bool lo  = lane < 16;
    const int koff = lo ? 0 : 2;               // K split across lane halves

    // Load A fragments once: row = tile*16 + l15, reused for all 512 k-tiles.
    const float* zrow = zn + (size_t)(tile * 16 + l15) * DIM + koff;
    v2f a[9];
    #pragma unroll
    for (int c = 0; c < 8; ++c) a[c] = *(const v2f*)(zrow + 4 * c);
    a[8].x = lo ? 1.0f : 0.0f;                 // augmented constant column
    a[8].y = 0.0f;

    const float hscale = lo ? -0.5f : 0.0f;    // hoisted lane-half predicate
    const v8f zeroC = { 0.f, 0.f, 0.f, 0.f, 0.f, 0.f, 0.f, 0.f };

    float bs[8];
    int   bi[8];
    #pragma unroll
    for (int v = 0; v < 8; ++v) { bs[v] = -3.0e38f; bi[v] = 0; }

    for (int kt = wave; kt < K_CODES / 16; kt += 8) {
        const int code = kt * 16 + l15;        // this lane's column / code id
        const float* erow = en + (size_t)code * DIM + koff;
        v2f b[8];
        #pragma unroll
        for (int c = 0; c < 8; ++c) b[c] = *(const v2f*)(erow + 4 * c);

        v2f b8;                                // augmented bias row (branch-free)
        b8.x = esq[code] * hscale;
        b8.y = 0.0f;

        // single-accumulator chain of 9 WMMAs: acc = A*B - 0.5*esq (per column)
        v8f acc = __builtin_amdgcn_wmma_f32_16x16x4_f32(
                      false, a[8], false, b8, (short)0, zeroC, false, false);
        #pragma unroll
        for (int c = 0; c < 8; ++c) {
            acc = __builtin_amdgcn_wmma_f32_16x16x4_f32(
                      false, a[c], false, b[c], (short)0, acc, false, false);
        }

        #pragma unroll
        for (int v = 0; v < 8; ++v) {
            float s = acc[v];                  // score = dot - esq/2
            if (s > bs[v]) { bs[v] = s; bi[v] = code; }  // strict > keeps first idx
        }
    }

    // Reduce over the 16 columns (lanes within each half-wave). Max with
    // index-min tie-break is associative+commutative -> butterfly is exact.
    #pragma unroll
    for (int v = 0; v < 8; ++v) {
        float s = bs[v]; int i = bi[v];
        #pragma unroll
        for (int m = 1; m <= 8; m <<= 1) {
            float os = __shfl_xor(s, m, 32);
            int   oi = __shfl_xor(i, m, 32);
            if (os > s || (os == s && oi < i)) { s = os; i = oi; }
        }
        bs[v] = s; bi[v] = i;
    }

    if (lane == 0) {            // rows 0..7 of the tile
        #pragma unroll
        for (int v = 0; v < 8; ++v) { sS[wave][v] = bs[v]; sI[wave][v] = bi[v]; }
    } else if (lane == 16) {    // rows 8..15 of the tile
        #pragma unroll
        for (int v = 0; v < 8; ++v) { sS[wave][8 + v] = bs[v]; sI[wave][8 + v] = bi[v]; }
    }
    __syncthreads();

    if (threadIdx.x < 16) {
        const int r = (int)threadIdx.x;
        float s = sS[0][r]; int i = sI[0][r];
        #pragma unroll
        for (int w = 1; w < 8; ++w) {
            float os = sS[w][r]; int oi = sI[w][r];
            if (os > s || (os == s && oi < i)) { s = os; i = oi; }
        }
        bestIdx[tile * 16 + r] = i;
    }
}

// ---------------------------------------------------------------------------
// Kernel 3: gather z_q = en[idx], write straight-through output and idx,
// produce per-block double partial sums of (z_q - zn)^2 (fixed order).
// One wave per row; 8 rows per block -> 2048 blocks.
// ---------------------------------------------------------------------------
__global__ __launch_bounds__(256) void k_finalize(const float* __restrict__ zn,
                                                  const float* __restrict__ en,
                                                  const int* __restrict__ bestIdx,
                                                  float* __restrict__ out_zq,
                                                  float* __restrict__ out_idx,
                                                  double* __restrict__ partial) {
    __shared__ double sp[8];
    const int warp = (int)(threadIdx.x >> 5);
    const int lane = (int)(threadIdx.x & 31);
    const int row  = (int)blockIdx.x * 8 + warp;

    const int idx = bestIdx[row];
    const float znv = zn[(size_t)row * DIM + lane];
    const float zq  = en[(size_t)idx * DIM + lane];   // already normalized
    out_zq[(size_t)row * DIM + lane] = znv + (zq - znv);  // straight-through value
    if (lane == 0) out_idx[row] = (float)idx;

    const float d = zq - znv;
    double ds = (double)d * (double)d;
    #pragma unroll
    for (int m = 16; m >= 1; m >>= 1) ds += __shfl_xor(ds, m, 32);
    if (lane == 0) sp[warp] = ds;
    __syncthreads();
    if (threadIdx.x == 0) {
        double acc = 0.0;
        #pragma unroll
        for (int w = 0; w < 8; ++w) acc += sp[w];
        partial[blockIdx.x] = acc;
    }
}

// ---------------------------------------------------------------------------
// Kernel 4: deterministic fixed-order reduction of 2048 partials -> loss.
// loss = (BETA + 1) * mean((z_q - zn)^2), BETA = 0.25.
// ---------------------------------------------------------------------------
__global__ __launch_bounds__(256) void k_loss(const double* __restrict__ partial,
                                              float* __restrict__ out_loss) {
    __shared__ double s[256];
    double acc = 0.0;
    for (int i = (int)threadIdx.x; i < 2048; i += 256) acc += partial[i];
    s[threadIdx.x] = acc;
    __syncthreads();
    for (int off = 128; off > 0; off >>= 1) {
        if ((int)threadIdx.x < off) s[threadIdx.x] += s[threadIdx.x + off];
        __syncthreads();
    }
    if (threadIdx.x == 0) {
        double mse = s[0] / (double)((size_t)N_ROWS * DIM);
        out_loss[0] = (float)(1.25 * mse);
    }
}

// ---------------------------------------------------------------------------
extern "C" void kernel_launch(void* const* d_in, const int* in_sizes, int n_in,
                              void* d_out, int out_size, void* d_ws, size_t ws_size,
                              hipStream_t stream) {
    (void)in_sizes; (void)n_in; (void)out_size; (void)ws_size;

    const float* z  = (const float*)d_in[0];   // [8,2048,32] fp32
    const float* cb = (const float*)d_in[1];   // [8192,32]   fp32
    float* out = (float*)d_out;                // 524288 zq_st | 16384 idx | 1 loss

    // Workspace layout (~3.3 MB)
    float*  zn      = (float*)d_ws;            // 524288 floats
    float*  en      = zn + (size_t)N_ROWS * DIM;      // 262144 floats
    float*  esq     = en + (size_t)K_CODES * DIM;     // 8192 floats
    int*    bidx    = (int*)(esq + K_CODES);          // 16384 ints
    double* partial = (double*)(bidx + N_ROWS);       // 2048 doubles (8B aligned)

    // 1) normalize both tensors: (16384 + 8192) rows, 8 waves/block
    k_normalize<<<(N_ROWS + K_CODES) / 8, 256, 0, stream>>>(z, cb, zn, en, esq);

    // 2) WMMA GEMM + argmax: one block per 16-row tile
    k_argmin<<<N_ROWS / 16, 256, 0, stream>>>(zn, en, esq, bidx);

    // 3) gather + straight-through output + loss partials
    k_finalize<<<N_ROWS / 8, 256, 0, stream>>>(zn, en, bidx,
                                               out,
                                               out + (size_t)N_ROWS * DIM,
                                               partial);

    // 4) final deterministic loss reduction
    k_loss<<<1, 256, 0, stream>>>(partial, out + (size_t)N_ROWS * DIM + N_ROWS);
}